// GraphAttention_29283087024999
// MI455X (gfx1250) — compile-verified
//
#include <hip/hip_runtime.h>
#include <math.h>

#define N_NODES  50000
#define N_EDGES  800000
#define DIM      128
#define HEADS    8
#define HEAD_DIM 16
#define NEG_SLOPE 0.2f
#define LDS_STRIDE 132   // 128 + 4 pad: rotates banks by 4 per row, keeps 16B alignment (132*4=528=33*16)

typedef float v2f __attribute__((ext_vector_type(2)));
typedef float v8f __attribute__((ext_vector_type(8)));

// ---------------------------------------------------------------------------
// WMMA f32 16x16x4 tile: D(16x16) = Atile(16x128) * W(ncol0..+16, 0..128)^T + bias
// A-frag: lane L -> row m=L&15, k-pair base 2*(L>>4); steps of K=4.
// B-frag: lane L -> col n=L&15 of output = row n of W; same k-pair select.
// C/D   : VGPR r -> row r + 8*(L>>4), col L&15.
// ---------------------------------------------------------------------------
__device__ __forceinline__ v8f wmma_tile_f32(const float* __restrict__ sA, int row0,
                                             const float* __restrict__ W, int wstride,
                                             int ncol0, float bias_val, int lane) {
    const int m  = lane & 15;
    const int kh = (lane >> 4) << 1;      // 0 or 2
    v8f acc;
#pragma unroll
    for (int r = 0; r < 8; ++r) acc[r] = bias_val;
    const float* arow = sA + (row0 + m) * LDS_STRIDE + kh;
    const float* brow = W + (size_t)(ncol0 + m) * wstride + kh;
#pragma unroll
    for (int s = 0; s < 32; ++s) {
        v2f a = *(const v2f*)(arow + 4 * s);
        v2f b = *(const v2f*)(brow + 4 * s);
        acc = __builtin_amdgcn_wmma_f32_16x16x4_f32(false, a, false, b, (short)0, acc,
                                                    false, false);
    }
    return acc;
}

__device__ __forceinline__ void store_tile(float* __restrict__ out, int grow0, int rowlimit,
                                           int col0, const v8f& acc, int lane) {
    const int n = lane & 15;
    const int rbase = (lane >> 4) * 8;
#pragma unroll
    for (int r = 0; r < 8; ++r) {
        int row = grow0 + rbase + r;
        if (row < rowlimit) out[(size_t)row * DIM + col0 + n] = acc[r];
    }
}

__device__ __forceinline__ void store_tile_lds(float* sOut, int row0, int col0,
                                               const v8f& acc, int lane) {
    const int n = lane & 15;
    const int rbase = (lane >> 4) * 8;
#pragma unroll
    for (int r = 0; r < 8; ++r)
        sOut[(row0 + rbase + r) * LDS_STRIDE + col0 + n] = acc[r];
}

// stage 64 rows x 128 floats global -> LDS (row index clamped for tail blocks)
__device__ __forceinline__ void load_tile_rows(const float* __restrict__ src, float* sDst,
                                               int grow0, int rowlimit, int tid) {
#pragma unroll
    for (int i = 0; i < 8; ++i) {
        int item = tid + i * 256;             // 0..2047
        int r  = item >> 5;                   // 0..63
        int c4 = item & 31;                   // 0..31 float4 within row
        int gr = grow0 + r;
        if (gr >= rowlimit) gr = rowlimit - 1;
        float4 v = ((const float4*)(src + (size_t)gr * DIM))[c4];
        *(float4*)(sDst + r * LDS_STRIDE + c4 * 4) = v;
    }
}

__device__ __forceinline__ void atomicMaxFloat(float* addr, float val) {
    if (val >= 0.0f) atomicMax((int*)addr, __float_as_int(val));
    else             atomicMin((unsigned int*)addr, __float_as_uint(val));
}

// ---------------------------------------------------------------------------
// K0: init workspace accumulators
// ---------------------------------------------------------------------------
__global__ void gat_init_kernel(float* __restrict__ segmax, float* __restrict__ denom,
                                float* __restrict__ agg) {
    int i = blockIdx.x * blockDim.x + threadIdx.x;
    if (i < N_NODES * HEADS) { segmax[i] = -__builtin_huge_valf(); denom[i] = 0.0f; }
    if (i < N_NODES * DIM)   agg[i] = 0.0f;
}

// ---------------------------------------------------------------------------
// K1: per-node projections  A = n@ni_w.T+ni_b, B = n@nj_w.T+nj_b, M = n@msg_wL.T+msg_b
// ---------------------------------------------------------------------------
__global__ void gat_node_proj_kernel(const float* __restrict__ nodes,
                                     const float* __restrict__ ni_w, const float* __restrict__ ni_b,
                                     const float* __restrict__ nj_w, const float* __restrict__ nj_b,
                                     const float* __restrict__ msg_w, const float* __restrict__ msg_b,
                                     float* __restrict__ Aout, float* __restrict__ Bout,
                                     float* __restrict__ Mout) {
    __shared__ float sA[64 * LDS_STRIDE];
    const int tid = threadIdx.x, lane = tid & 31, wave = tid >> 5;
    const int grow0 = blockIdx.x * 64;
    load_tile_rows(nodes, sA, grow0, N_NODES, tid);
    __syncthreads();
    for (int t = wave; t < 96; t += 8) {       // 4 row groups x 24 col tiles
        int rg = t / 24, ct = t % 24;
        const float *W, *bias; float* out; int wstride, oct;
        if (ct < 8)       { W = ni_w;  bias = ni_b;  out = Aout; wstride = DIM;     oct = ct; }
        else if (ct < 16) { W = nj_w;  bias = nj_b;  out = Bout; wstride = DIM;     oct = ct - 8; }
        else              { W = msg_w; bias = msg_b; out = Mout; wstride = 2 * DIM; oct = ct - 16; }
        int ncol0 = oct * 16;
        float bv = bias[ncol0 + (lane & 15)];
        v8f acc = wmma_tile_f32(sA, rg * 16, W, wstride, ncol0, bv, lane);
        store_tile(out, grow0 + rg * 16, N_NODES, ncol0, acc, lane);
    }
}

// ---------------------------------------------------------------------------
// K2: per-edge: eproj = ef@e_w.T+e_b (LDS), MEproj = ef@msg_wR.T (global),
//     then fused leaky-relu + per-head logits + segment-max atomics.
// ---------------------------------------------------------------------------
__global__ void gat_edge_kernel(const float* __restrict__ ef,
                                const float* __restrict__ e_w, const float* __restrict__ e_b,
                                const float* __restrict__ msg_w,
                                const int* __restrict__ src, const int* __restrict__ dst,
                                const float* __restrict__ Aout, const float* __restrict__ Bout,
                                const float* __restrict__ attn_proj,
                                float* __restrict__ MEout, float* __restrict__ Lout,
                                float* __restrict__ segmax) {
    __shared__ float sE[64 * LDS_STRIDE];
    __shared__ float sH[64 * LDS_STRIDE];
    const int tid = threadIdx.x, lane = tid & 31, wave = tid >> 5;
    const int ge0 = blockIdx.x * 64;
    load_tile_rows(ef, sE, ge0, N_EDGES, tid);
    __syncthreads();
    for (int t = wave; t < 64; t += 8) {       // 4 row groups x 16 col tiles
        int rg = t / 16, ct = t % 16;
        if (ct < 8) {
            int ncol0 = ct * 16;
            float bv = e_b[ncol0 + (lane & 15)];
            v8f acc = wmma_tile_f32(sE, rg * 16, e_w, DIM, ncol0, bv, lane);
            store_tile_lds(sH, rg * 16, ncol0, acc, lane);
        } else {
            int ncol0 = (ct - 8) * 16;
            v8f acc = wmma_tile_f32(sE, rg * 16, msg_w + DIM, 2 * DIM, ncol0, 0.0f, lane);
            store_tile(MEout, ge0 + rg * 16, N_EDGES, ncol0, acc, lane);
        }
    }
    __syncthreads();
    // phase 2: 64 edges x 8 heads = 512 tasks over 256 threads
    for (int task = tid; task < 512; task += 256) {
        int le = task >> 3, h = task & 7;
        int ge = ge0 + le;
        int s = src[ge], d = dst[ge];
        const float4* ap = (const float4*)(Aout + (size_t)s * DIM + h * HEAD_DIM);
        const float4* bp = (const float4*)(Bout + (size_t)d * DIM + h * HEAD_DIM);
        const float4* pp = (const float4*)(attn_proj + h * HEAD_DIM);
        const float*  hp = sH + le * LDS_STRIDE + h * HEAD_DIM;
        float logit = 0.0f;
#pragma unroll
        for (int q = 0; q < 4; ++q) {
            float4 av = ap[q], bv = bp[q], pv = pp[q];
            float4 hv = *(const float4*)(hp + q * 4);
            float x;
            x = av.x + bv.x + hv.x; x = (x >= 0.0f) ? x : NEG_SLOPE * x; logit += x * pv.x;
            x = av.y + bv.y + hv.y; x = (x >= 0.0f) ? x : NEG_SLOPE * x; logit += x * pv.y;
            x = av.z + bv.z + hv.z; x = (x >= 0.0f) ? x : NEG_SLOPE * x; logit += x * pv.z;
            x = av.w + bv.w + hv.w; x = (x >= 0.0f) ? x : NEG_SLOPE * x; logit += x * pv.w;
        }
        Lout[(size_t)ge * HEADS + h] = logit;
        atomicMaxFloat(segmax + (size_t)s * HEADS + h, logit);
    }
}

// ---------------------------------------------------------------------------
// K3: ex = exp(logit - segmax[src]); denom += ex   (L overwritten with ex)
// ---------------------------------------------------------------------------
__global__ void gat_softmax_num_kernel(const int* __restrict__ src,
                                       const float* __restrict__ segmax,
                                       float* __restrict__ L, float* __restrict__ denom) {
    int i = blockIdx.x * blockDim.x + threadIdx.x;
    if (i >= N_EDGES * HEADS) return;
    int e = i >> 3, h = i & 7;
    int s = src[e];
    float ex = expf(L[i] - segmax[(size_t)s * HEADS + h]);
    L[i] = ex;
    atomicAdd(denom + (size_t)s * HEADS + h, ex);
}

// ---------------------------------------------------------------------------
// K4: agg[src] += (M[dst] + MEproj[e]) * (ex / denom[src])  per head channel
//     task = (edge, float4-chunk of 128 channels) -> E*32 tasks
// ---------------------------------------------------------------------------
__global__ void gat_scatter_kernel(const int* __restrict__ src, const int* __restrict__ dst,
                                   const float* __restrict__ L, const float* __restrict__ denom,
                                   const float* __restrict__ M, const float* __restrict__ ME,
                                   float* __restrict__ agg) {
    int i = blockIdx.x * blockDim.x + threadIdx.x;
    if (i >= N_EDGES * 32) return;
    int e = i >> 5, q = i & 31;
    int h = q >> 2;
    int s = src[e], d = dst[e];
    float attn = L[(size_t)e * HEADS + h] / denom[(size_t)s * HEADS + h];
    float4 me = ((const float4*)(ME + (size_t)e * DIM))[q];
    float4 mm = ((const float4*)(M  + (size_t)d * DIM))[q];
    float* ap = agg + (size_t)s * DIM + q * 4;
    atomicAdd(ap + 0, (me.x + mm.x) * attn);
    atomicAdd(ap + 1, (me.y + mm.y) * attn);
    atomicAdd(ap + 2, (me.z + mm.z) * attn);
    atomicAdd(ap + 3, (me.w + mm.w) * attn);
}

// ---------------------------------------------------------------------------
// K5: out = agg @ out_w.T + out_b
// ---------------------------------------------------------------------------
__global__ void gat_out_proj_kernel(const float* __restrict__ agg,
                                    const float* __restrict__ out_w,
                                    const float* __restrict__ out_b,
                                    float* __restrict__ out) {
    __shared__ float sA[64 * LDS_STRIDE];
    const int tid = threadIdx.x, lane = tid & 31, wave = tid >> 5;
    const int grow0 = blockIdx.x * 64;
    load_tile_rows(agg, sA, grow0, N_NODES, tid);
    __syncthreads();
    for (int t = wave; t < 32; t += 8) {       // 4 row groups x 8 col tiles
        int rg = t / 8, ct = t % 8;
        int ncol0 = ct * 16;
        float bv = out_b[ncol0 + (lane & 15)];
        v8f acc = wmma_tile_f32(sA, rg * 16, out_w, DIM, ncol0, bv, lane);
        store_tile(out, grow0 + rg * 16, N_NODES, ncol0, acc, lane);
    }
}

// ---------------------------------------------------------------------------
extern "C" void kernel_launch(void* const* d_in, const int* in_sizes, int n_in,
                              void* d_out, int out_size, void* d_ws, size_t ws_size,
                              hipStream_t stream) {
    (void)in_sizes; (void)n_in; (void)out_size; (void)ws_size;
    const float* nodes     = (const float*)d_in[0];
    const float* edgef     = (const float*)d_in[1];
    const int*   src       = (const int*)d_in[2];
    const int*   dst       = (const int*)d_in[3];
    const float* ni_w      = (const float*)d_in[4];
    const float* ni_b      = (const float*)d_in[5];
    const float* nj_w      = (const float*)d_in[6];
    const float* nj_b      = (const float*)d_in[7];
    const float* e_w       = (const float*)d_in[8];
    const float* e_b       = (const float*)d_in[9];
    const float* attn_proj = (const float*)d_in[10];
    const float* msg_w     = (const float*)d_in[11];
    const float* msg_b     = (const float*)d_in[12];
    const float* out_w     = (const float*)d_in[13];
    const float* out_b     = (const float*)d_in[14];
    float* out = (float*)d_out;

    // workspace layout
    char* ws = (char*)d_ws;
    const size_t NA = (size_t)N_NODES * DIM * sizeof(float);     // 25.6 MB
    const size_t NE = (size_t)N_EDGES * DIM * sizeof(float);     // 409.6 MB
    const size_t NL = (size_t)N_EDGES * HEADS * sizeof(float);   // 25.6 MB
    const size_t NH = (size_t)N_NODES * HEADS * sizeof(float);   // 1.6 MB
    float* A      = (float*)(ws);               // node @ ni_w.T + ni_b
    float* B      = (float*)(ws + NA);          // node @ nj_w.T + nj_b
    float* M      = (float*)(ws + 2 * NA);      // node @ msg_wL.T + msg_b
    float* agg    = (float*)(ws + 3 * NA);      // scatter accumulator
    float* ME     = (float*)(ws + 4 * NA);      // ef @ msg_wR.T
    float* L      = (float*)(ws + 4 * NA + NE); // logits, then ex
    float* segmax = (float*)(ws + 4 * NA + NE + NL);
    float* denom  = (float*)(ws + 4 * NA + NE + NL + NH);

    const int node_blocks = (N_NODES + 63) / 64;     // 782
    const int edge_blocks = N_EDGES / 64;            // 12500

    gat_init_kernel<<<(N_NODES * DIM + 255) / 256, 256, 0, stream>>>(segmax, denom, agg);
    gat_node_proj_kernel<<<node_blocks, 256, 0, stream>>>(nodes, ni_w, ni_b, nj_w, nj_b,
                                                          msg_w, msg_b, A, B, M);
    gat_edge_kernel<<<edge_blocks, 256, 0, stream>>>(edgef, e_w, e_b, msg_w, src, dst,
                                                     A, B, attn_proj, ME, L, segmax);
    gat_softmax_num_kernel<<<(N_EDGES * HEADS + 255) / 256, 256, 0, stream>>>(src, segmax, L, denom);
    gat_scatter_kernel<<<(N_EDGES * 32 + 255) / 256, 256, 0, stream>>>(src, dst, L, denom, M, ME, agg);
    gat_out_proj_kernel<<<node_blocks, 256, 0, stream>>>(agg, out_w, out_b, out);
}